// GCN_66589172957447
// MI455X (gfx1250) — compile-verified
//
#include <hip/hip_runtime.h>

typedef __attribute__((ext_vector_type(2))) float v2f;
typedef __attribute__((ext_vector_type(8))) float v8f;

#define IN_CH  512
#define HID    16
#define OUT_CH 64
#define THREADS 256

// ---------------- degree / norm ----------------
__global__ void k_init_deg(float* __restrict__ deg, int n) {
  int i = blockIdx.x * blockDim.x + threadIdx.x;
  if (i < n) deg[i] = 1.0f;  // self-loop contributes 1
}

__global__ void k_count_deg(const int* __restrict__ dst, float* __restrict__ deg, int e) {
  int i = blockIdx.x * blockDim.x + threadIdx.x;
  if (i < e) atomicAdd(&deg[dst[i]], 1.0f);
}

__global__ void k_rsqrt_inplace(float* __restrict__ deg, int n) {
  int i = blockIdx.x * blockDim.x + threadIdx.x;
  if (i < n) deg[i] = rsqrtf(deg[i]);  // deg >= 1 always
}

__global__ void k_edge_norm(const int* __restrict__ src, const int* __restrict__ dst,
                            const float* __restrict__ dinv, float* __restrict__ norm, int e) {
  int i = blockIdx.x * blockDim.x + threadIdx.x;
  if (i < e) norm[i] = dinv[src[i]] * dinv[dst[i]];
}

// ---------------- GEMM1: x[N,512] @ W1[512,16] -> xw1[N,16] ----------------
// One wave computes a 16x16 output tile using V_WMMA_F32_16X16X4_F32, K-loop of 128 steps.
__global__ void k_gemm1(const float* __restrict__ x, const float* __restrict__ W1,
                        float* __restrict__ xw1, int nTiles) {
  __shared__ float w1s[IN_CH * HID];  // 32 KB
  for (int i = threadIdx.x; i < IN_CH * HID; i += blockDim.x) w1s[i] = W1[i];
  __syncthreads();

  int wave = threadIdx.x >> 5;
  int lane = threadIdx.x & 31;
  int t = blockIdx.x * (blockDim.x >> 5) + wave;
  if (t >= nTiles) return;  // uniform across wave: EXEC stays all-1s for WMMA

  int half = lane >> 4;   // 0: K={0,1}, 1: K={2,3} within each 4-wide K step
  int ml = lane & 15;     // M (for A) / N (for B)
  const float* xr = x + (size_t)(t * 16 + ml) * IN_CH + half * 2;

  v8f c = {};
  for (int kk = 0; kk < IN_CH / 4; ++kk) {
    v2f a = *(const v2f*)(xr + kk * 4);           // A[M=ml][k0], A[M=ml][k0+1]
    int k0 = kk * 4 + half * 2;
    v2f b;
    b.x = w1s[k0 * HID + ml];                     // B[K=k0][N=ml]
    b.y = w1s[(k0 + 1) * HID + ml];               // B[K=k0+1][N=ml]
    c = __builtin_amdgcn_wmma_f32_16x16x4_f32(false, a, false, b, (short)0, c, false, false);
  }

  float* o = xw1 + (size_t)(t * 16) * HID + ml;
  #pragma unroll
  for (int r = 0; r < 8; ++r)
    o[(half * 8 + r) * HID] = c[r];               // VGPR r: M=r (lanes<16) / M=8+r
}

// ---------------- GEMM2: h[N,16] @ W2[16,64] -> h2[N,64] ----------------
__global__ void k_gemm2(const float* __restrict__ h, const float* __restrict__ W2,
                        float* __restrict__ h2, int nTiles) {
  __shared__ float w2s[HID * OUT_CH];  // 4 KB
  for (int i = threadIdx.x; i < HID * OUT_CH; i += blockDim.x) w2s[i] = W2[i];
  __syncthreads();

  int wave = threadIdx.x >> 5;
  int lane = threadIdx.x & 31;
  int t = blockIdx.x * (blockDim.x >> 5) + wave;
  if (t >= nTiles) return;

  int half = lane >> 4;
  int ml = lane & 15;
  const float* hr = h + (size_t)(t * 16 + ml) * HID + half * 2;

  v8f c0 = {}, c1 = {}, c2 = {}, c3 = {};
  #pragma unroll
  for (int kk = 0; kk < HID / 4; ++kk) {
    v2f a = *(const v2f*)(hr + kk * 4);
    int k0 = kk * 4 + half * 2;
    v2f b0, b1, b2, b3;
    b0.x = w2s[k0 * OUT_CH +  0 + ml]; b0.y = w2s[(k0 + 1) * OUT_CH +  0 + ml];
    b1.x = w2s[k0 * OUT_CH + 16 + ml]; b1.y = w2s[(k0 + 1) * OUT_CH + 16 + ml];
    b2.x = w2s[k0 * OUT_CH + 32 + ml]; b2.y = w2s[(k0 + 1) * OUT_CH + 32 + ml];
    b3.x = w2s[k0 * OUT_CH + 48 + ml]; b3.y = w2s[(k0 + 1) * OUT_CH + 48 + ml];
    c0 = __builtin_amdgcn_wmma_f32_16x16x4_f32(false, a, false, b0, (short)0, c0, false, false);
    c1 = __builtin_amdgcn_wmma_f32_16x16x4_f32(false, a, false, b1, (short)0, c1, false, false);
    c2 = __builtin_amdgcn_wmma_f32_16x16x4_f32(false, a, false, b2, (short)0, c2, false, false);
    c3 = __builtin_amdgcn_wmma_f32_16x16x4_f32(false, a, false, b3, (short)0, c3, false, false);
  }

  float* o = h2 + (size_t)(t * 16) * OUT_CH + ml;
  #pragma unroll
  for (int r = 0; r < 8; ++r) {
    int row = half * 8 + r;
    o[row * OUT_CH +  0] = c0[r];
    o[row * OUT_CH + 16] = c1[r];
    o[row * OUT_CH + 32] = c2[r];
    o[row * OUT_CH + 48] = c3[r];
  }
}

// ---------------- aggregation ----------------
// out[n*C+c] = bias[c] + in[n*C+c] * dinv[n]^2   (self-loop term, also clears poison)
template <int LOGC>
__global__ void k_init_selfloop(const float* __restrict__ in, const float* __restrict__ dinv,
                                const float* __restrict__ bias, float* __restrict__ out,
                                long long total) {
  long long i = (long long)blockIdx.x * blockDim.x + threadIdx.x;
  if (i >= total) return;
  int node = (int)(i >> LOGC);
  int ch = (int)(i & ((1 << LOGC) - 1));
  float di = dinv[node];
  out[i] = bias[ch] + in[i] * di * di;
}

// out[dst*C+c] += in[src*C+c] * norm[e]
template <int LOGC>
__global__ void k_agg_edges(const int* __restrict__ src, const int* __restrict__ dst,
                            const float* __restrict__ norm, const float* __restrict__ in,
                            float* __restrict__ out, long long total) {
  long long i = (long long)blockIdx.x * blockDim.x + threadIdx.x;
  if (i >= total) return;
  int e = (int)(i >> LOGC);
  int ch = (int)(i & ((1 << LOGC) - 1));
  int s = src[e], d = dst[e];
  float v = in[((size_t)s << LOGC) + ch] * norm[e];
  atomicAdd(&out[((size_t)d << LOGC) + ch], v);
}

__global__ void k_relu(float* __restrict__ h, long long total) {
  long long i = (long long)blockIdx.x * blockDim.x + threadIdx.x;
  if (i < total) h[i] = fmaxf(h[i], 0.0f);
}

static inline int gridFor(long long total, int threads) {
  return (int)((total + threads - 1) / threads);
}

extern "C" void kernel_launch(void* const* d_in, const int* in_sizes, int n_in,
                              void* d_out, int out_size, void* d_ws, size_t ws_size,
                              hipStream_t stream) {
  const float* x  = (const float*)d_in[0];
  const int* ei   = (const int*)d_in[1];
  const float* W1 = (const float*)d_in[2];
  const float* b1 = (const float*)d_in[3];
  const float* W2 = (const float*)d_in[4];
  const float* b2 = (const float*)d_in[5];
  float* out = (float*)d_out;

  const int n = in_sizes[0] / IN_CH;      // 100000
  const int e = in_sizes[1] / 2;          // 3200000
  const int* src = ei;
  const int* dst = ei + e;

  // workspace layout (floats)
  float* dinv = (float*)d_ws;             // n  (degree, then rsqrt in place)
  float* norm = dinv + n;                 // e
  float* xw1  = norm + e;                 // n*HID
  float* h    = xw1 + (size_t)n * HID;    // n*HID
  float* h2   = h   + (size_t)n * HID;    // n*OUT_CH

  const int nTiles = (n + 15) / 16;
  const int wavesPerBlock = THREADS / 32;

  // degrees + normalization
  k_init_deg<<<gridFor(n, THREADS), THREADS, 0, stream>>>(dinv, n);
  k_count_deg<<<gridFor(e, THREADS), THREADS, 0, stream>>>(dst, dinv, e);
  k_rsqrt_inplace<<<gridFor(n, THREADS), THREADS, 0, stream>>>(dinv, n);
  k_edge_norm<<<gridFor(e, THREADS), THREADS, 0, stream>>>(src, dst, dinv, norm, e);

  // layer 1
  k_gemm1<<<gridFor(nTiles, wavesPerBlock), THREADS, 0, stream>>>(x, W1, xw1, nTiles);
  long long tot1 = (long long)n * HID;
  k_init_selfloop<4><<<gridFor(tot1, THREADS), THREADS, 0, stream>>>(xw1, dinv, b1, h, tot1);
  long long te1 = (long long)e * HID;
  k_agg_edges<4><<<gridFor(te1, THREADS), THREADS, 0, stream>>>(src, dst, norm, xw1, h, te1);
  k_relu<<<gridFor(tot1, THREADS), THREADS, 0, stream>>>(h, tot1);

  // layer 2
  k_gemm2<<<gridFor(nTiles, wavesPerBlock), THREADS, 0, stream>>>(h, W2, h2, nTiles);
  long long tot2 = (long long)n * OUT_CH;
  k_init_selfloop<6><<<gridFor(tot2, THREADS), THREADS, 0, stream>>>(h2, dinv, b2, out, tot2);
  long long te2 = (long long)e * OUT_CH;
  k_agg_edges<6><<<gridFor(te2, THREADS), THREADS, 0, stream>>>(src, dst, norm, h2, out, te2);
}